// MonarchLinearLayer_7584912245165
// MI455X (gfx1250) — compile-verified
//
#include <hip/hip_runtime.h>
#include <hip/hip_bf16.h>
#include <cstdint>

// Monarch diagonal-butterfly chain collapses to one gather+scale:
//   out[row, j] = x[row, g[j]] * t[j]
// where (g, t) are the composed permutation/scale over the 11 stages.
// Memory-bound (512 MB traffic @ 23.3 TB/s ~ 22 us). CDNA5 paths used:
//   - TDM tensor_load_to_lds for the (g,t) tables + s_wait_tensorcnt
//   - non-temporal b128 global streaming (TH=NT) to keep the 192MB L2 clean
//   - global_prefetch for the next row
//   - wave32-sized gating of the TDM issue

#define DIM   2048
#define NB    11
#define TPB   256
#define ROWS_PER_BLOCK 8

typedef __attribute__((ext_vector_type(4))) unsigned int u32x4;
typedef __attribute__((ext_vector_type(4))) int i32x4;
typedef __attribute__((ext_vector_type(8))) int i32x8;
typedef __attribute__((ext_vector_type(4))) float f32x4;   // true clang vector:
                                                           // valid for nontemporal builtins

#if defined(__has_builtin)
#if __has_builtin(__builtin_amdgcn_tensor_load_to_lds) && __has_builtin(__builtin_amdgcn_s_wait_tensorcnt)
#define MONARCH_USE_TDM 1
#endif
#endif

// ---------------------------------------------------------------------------
// Kernel 1: compose the 11 (perm, diag) stages into g[DIM], t[DIM] + identity
// flag. One workgroup; each thread traces 2 output indices backward.
// ---------------------------------------------------------------------------
__global__ __launch_bounds__(1024)
void monarch_compose(const float* __restrict__ blocks,
                     const int* __restrict__ perms,
                     int* __restrict__ g,
                     float* __restrict__ t,
                     int* __restrict__ identFlag) {
  __shared__ int s_notIdent;
  if (threadIdx.x == 0) s_notIdent = 0;
  __syncthreads();

  int local = 0;
  for (int j = (int)threadIdx.x; j < DIM; j += 1024) {
    int idx = j;
    // last stage has no permutation after it
    float total = blocks[(NB - 1) * DIM + idx];
#pragma unroll
    for (int i = NB - 2; i >= 0; --i) {
      idx = perms[i * DIM + idx];
      total *= blocks[i * DIM + idx];
    }
    g[j] = idx;
    t[j] = total;
    if (idx != j) local = 1;
  }
  if (local) atomicOr(&s_notIdent, 1);
  __syncthreads();
  if (threadIdx.x == 0) identFlag[0] = s_notIdent ? 0 : 1;
}

// ---------------------------------------------------------------------------
// TDM helper: 1-row (2D degenerate) descriptor moving n_elems f32 from global
// to LDS. Descriptor layout per CDNA5 ISA ch.8 (D# groups 0/1).
// ---------------------------------------------------------------------------
#ifdef MONARCH_USE_TDM
__device__ __forceinline__ void tdm_load_row_b32(uint32_t lds_addr,
                                                 const void* src,
                                                 uint32_t n_elems) {
  uint64_t ga = (uint64_t)(uintptr_t)src;
  u32x4 g0;
  g0[0] = 1u;                                           // count=1 (valid), user mode
  g0[1] = lds_addr;                                     // LDS byte address
  g0[2] = (uint32_t)ga;                                 // global_addr[31:0]
  g0[3] = (uint32_t)((ga >> 32) & 0x01FFFFFFu)          // global_addr[56:32]
        | (2u << 30);                                   // type = 2 ("image")
  i32x8 g1;
  g1[0] = (int)(2u << 16);                              // data_size = 4 B
  g1[1] = (int)(n_elems << 16);                         // tensor_dim0[15:0]
  g1[2] = (int)((n_elems >> 16) | (1u << 16));          // tensor_dim0[31:16], tensor_dim1=1
  g1[3] = (int)(n_elems << 16);                         // tile_dim0
  g1[4] = 1;                                            // tile_dim1 = 1
  g1[5] = (int)n_elems;                                 // tensor_dim0_stride[31:0]
  g1[6] = 0;                                            // stride hi / dim1_stride lo
  g1[7] = 0;
  i32x4 z4 = {0, 0, 0, 0};
#if defined(__clang_major__) && __clang_major__ >= 23
  i32x8 z8 = {0, 0, 0, 0, 0, 0, 0, 0};
  __builtin_amdgcn_tensor_load_to_lds(g0, g1, z4, z4, z8, 0);
#else
  __builtin_amdgcn_tensor_load_to_lds(g0, g1, z4, z4, 0);
#endif
}
#endif

// ---------------------------------------------------------------------------
// Kernel 2: streaming gather+scale. 8 rows (8 KB each) per workgroup.
// x/out are touched exactly once -> non-temporal hints keep L2 for the tables.
// ---------------------------------------------------------------------------
__global__ __launch_bounds__(TPB)
void monarch_apply(const float* __restrict__ x,
                   float* __restrict__ out,
                   const int* __restrict__ g,
                   const float* __restrict__ t,
                   const int* __restrict__ identFlag) {
  __shared__ __align__(16) float sh_t[DIM];
  __shared__ __align__(16) int   sh_g[DIM];

#ifdef MONARCH_USE_TDM
  // Wave 0 issues two TDM descriptors (t then g) and waits on TENSORcnt;
  // the workgroup barrier releases the other waves once LDS is populated.
  if (threadIdx.x < 32) {
    uint32_t lds_t = (uint32_t)(uintptr_t)&sh_t[0];   // low 32 bits = LDS offset
    uint32_t lds_g = (uint32_t)(uintptr_t)&sh_g[0];
    if (threadIdx.x == 0) {
      tdm_load_row_b32(lds_t, t, DIM);
      tdm_load_row_b32(lds_g, g, DIM);
    }
    __builtin_amdgcn_s_wait_tensorcnt(0);
  }
  __syncthreads();
#else
  for (int j = (int)threadIdx.x; j < DIM; j += TPB) {
    sh_t[j] = t[j];
    sh_g[j] = g[j];
  }
  __syncthreads();
#endif

  const int row0 = (int)blockIdx.x * ROWS_PER_BLOCK;
  const int ident = identFlag[0];  // uniform scalar load -> uniform branch

  if (ident) {
    // Fast path: permutation composes to identity -> pure NT b128 streaming.
    const f32x4* __restrict__ t4 = (const f32x4*)sh_t;
#pragma unroll 2
    for (int r = 0; r < ROWS_PER_BLOCK; ++r) {
      const size_t base = (size_t)(row0 + r) * DIM;
      const f32x4* __restrict__ xr = (const f32x4*)(x + base);
      f32x4* __restrict__ orow = (f32x4*)(out + base);
#pragma unroll
      for (int it = 0; it < DIM / 4 / TPB; ++it) {      // 2 iters of 256 lanes
        const int j = it * TPB + (int)threadIdx.x;
        if (r + 1 < ROWS_PER_BLOCK) {
          __builtin_prefetch(xr + (DIM / 4) + j, 0, 0); // next row, same lane
        }
        f32x4 v = __builtin_nontemporal_load(xr + j);   // global_load_b128 NT
        const f32x4 s = t4[j];                          // ds_load_b128 (reused)
        v *= s;
        __builtin_nontemporal_store(v, orow + j);       // global_store_b128 NT
      }
    }
  } else {
    // General path: per-element gather through LDS tables, NT b128 stores.
    for (int r = 0; r < ROWS_PER_BLOCK; ++r) {
      const size_t base = (size_t)(row0 + r) * DIM;
#pragma unroll
      for (int it = 0; it < DIM / 4 / TPB; ++it) {
        const int j4 = (it * TPB + (int)threadIdx.x) * 4;
        f32x4 v;
        v.x = x[base + (size_t)sh_g[j4 + 0]] * sh_t[j4 + 0];
        v.y = x[base + (size_t)sh_g[j4 + 1]] * sh_t[j4 + 1];
        v.z = x[base + (size_t)sh_g[j4 + 2]] * sh_t[j4 + 2];
        v.w = x[base + (size_t)sh_g[j4 + 3]] * sh_t[j4 + 3];
        __builtin_nontemporal_store(v, (f32x4*)(out + base + j4));
      }
    }
  }
}

// ---------------------------------------------------------------------------
extern "C" void kernel_launch(void* const* d_in, const int* in_sizes, int n_in,
                              void* d_out, int out_size, void* d_ws, size_t ws_size,
                              hipStream_t stream) {
  const float* x      = (const float*)d_in[0];  // (4, 8192, 2048) f32
  const float* blocks = (const float*)d_in[1];  // (11, 2, 1024) f32
  const int*   perms  = (const int*)d_in[2];    // (11, 2048) i32

  float* out = (float*)d_out;

  // Workspace layout: g[DIM] int | t[DIM] float | identFlag int
  int*   g    = (int*)d_ws;
  float* t    = (float*)((char*)d_ws + DIM * sizeof(int));
  int*   flag = (int*)((char*)d_ws + 2 * DIM * sizeof(int));

  hipLaunchKernelGGL(monarch_compose, dim3(1), dim3(1024), 0, stream,
                     blocks, perms, g, t, flag);

  const int rows = out_size / DIM;               // 32768
  const int grid = rows / ROWS_PER_BLOCK;        // 4096
  hipLaunchKernelGGL(monarch_apply, dim3(grid), dim3(TPB), 0, stream,
                     x, out, g, t, flag);
}